// DynamicSimpleVFE_26834955665464
// MI455X (gfx1250) — compile-verified
//
#include <hip/hip_runtime.h>
#include <stdint.h>

#define NPTS   2000000
#define CCH    16
#define CMAX   128
#define KSPACE (CMAX*CMAX*CMAX)        /* 2,097,152 = 2^21 */
#define SCAN_TPB 256
#define SCAN_EPT 8
#define SCAN_EPB (SCAN_TPB*SCAN_EPT)   /* 2048 */
#define SCAN_NBLK (KSPACE/SCAN_EPB)    /* 1024 */

typedef float v2f __attribute__((ext_vector_type(2)));
typedef float v8f __attribute__((ext_vector_type(8)));

// ---------------------------------------------------------------- pass 1: dense per-key counts
__global__ void vfe_count(const int* __restrict__ coors, uint32_t* __restrict__ cnt) {
    int i = blockIdx.x * blockDim.x + threadIdx.x;
    if (i >= NPTS) return;
    int c0 = coors[3*i], c1 = coors[3*i+1], c2 = coors[3*i+2];
    uint32_t key = ((uint32_t)c0 * CMAX + (uint32_t)c1) * CMAX + (uint32_t)c2;
    atomicAdd(&cnt[key], 1u);
}

// ---------------------------------------------------------------- pass 2a: block-local exclusive scan of occupancy flags
__global__ void vfe_scan1(const uint32_t* __restrict__ cnt, uint32_t* __restrict__ rank,
                          uint32_t* __restrict__ bsum) {
    __shared__ uint32_t sdata[SCAN_TPB];
    int tid = threadIdx.x;
    uint32_t base = (uint32_t)blockIdx.x * SCAN_EPB + (uint32_t)tid * SCAN_EPT;
    uint32_t f[SCAN_EPT]; uint32_t s = 0;
    #pragma unroll
    for (int j = 0; j < SCAN_EPT; ++j) { f[j] = cnt[base + j] ? 1u : 0u; s += f[j]; }
    sdata[tid] = s;
    __syncthreads();
    for (int off = 1; off < SCAN_TPB; off <<= 1) {
        uint32_t v = (tid >= off) ? sdata[tid - off] : 0u;
        __syncthreads();
        sdata[tid] += v;
        __syncthreads();
    }
    uint32_t run = sdata[tid] - s;   // exclusive prefix within block
    #pragma unroll
    for (int j = 0; j < SCAN_EPT; ++j) { rank[base + j] = run; run += f[j]; }
    if (tid == SCAN_TPB - 1) bsum[blockIdx.x] = sdata[tid];
}

// ---------------------------------------------------------------- pass 2b: scan the 1024 block sums, emit total M
__global__ void vfe_scan2(const uint32_t* __restrict__ bsum, uint32_t* __restrict__ boff,
                          uint32_t* __restrict__ mtot) {
    __shared__ uint32_t sdata[SCAN_NBLK];
    int tid = threadIdx.x;              // launched with SCAN_NBLK = 1024 threads
    uint32_t v0 = bsum[tid];
    sdata[tid] = v0;
    __syncthreads();
    for (int off = 1; off < SCAN_NBLK; off <<= 1) {
        uint32_t v = (tid >= off) ? sdata[tid - off] : 0u;
        __syncthreads();
        sdata[tid] += v;
        __syncthreads();
    }
    boff[tid] = sdata[tid] - v0;
    if (tid == SCAN_NBLK - 1) *mtot = sdata[tid];
}

// ---------------------------------------------------------------- pass 2c: finalize ranks, write sorted unique key list
__global__ void vfe_compact(const uint32_t* __restrict__ cnt, uint32_t* __restrict__ rank,
                            const uint32_t* __restrict__ boff, uint32_t* __restrict__ uniq) {
    uint32_t k = blockIdx.x * blockDim.x + threadIdx.x;
    if (k >= KSPACE) return;
    if (cnt[k]) {
        uint32_t slot = rank[k] + boff[k >> 11];   // 2048 = 2^11 elems per scan block
        rank[k] = slot;
        uniq[slot] = k;
    }
}

// ---------------------------------------------------------------- pass 3: scatter-add feature sums into d_out
__global__ void vfe_scatter(const float* __restrict__ features, const int* __restrict__ coors,
                            const uint32_t* __restrict__ rank, float* __restrict__ outF) {
    uint32_t idx = blockIdx.x * blockDim.x + threadIdx.x;   // N*4 threads: (point, quad-channel)
    if (idx >= (uint32_t)NPTS * 4u) return;
    uint32_t i = idx >> 2, q = idx & 3u;
    int c0 = coors[3*i], c1 = coors[3*i+1], c2 = coors[3*i+2];
    uint32_t key = ((uint32_t)c0 * CMAX + (uint32_t)c1) * CMAX + (uint32_t)c2;
    uint32_t slot = rank[key];
    const float4 f = *(const float4*)(features + (size_t)i * CCH + q * 4u);
    float* dst = outF + (size_t)slot * CCH + q * 4u;
    atomicAdd(dst + 0, f.x);
    atomicAdd(dst + 1, f.y);
    atomicAdd(dst + 2, f.z);
    atomicAdd(dst + 3, f.w);
}

// ---------------------------------------------------------------- pass 4: WMMA row-scaling finalize + coords
// Per 16-voxel x 16-channel tile: D = diag(1/count) * S, computed as four
// chained V_WMMA_F32_16X16X4_F32 (A_q = 16x4 diagonal chunk, B_q = 4x16 row
// chunk of the sums tile). Padded slots (>= M) get scale 0 -> zeros.
__global__ void vfe_finalize(float* __restrict__ outF, int* __restrict__ outC,
                             const uint32_t* __restrict__ cnt, const uint32_t* __restrict__ uniq,
                             const uint32_t* __restrict__ mtot) {
    const uint32_t M = *mtot;
    int lane = threadIdx.x & 31;
    int wave = threadIdx.x >> 5;
    uint32_t tile = (uint32_t)blockIdx.x * 4u + (uint32_t)wave;   // 4 waves per 128-thread block
    uint32_t s0 = tile * 16u;
    int nIdx = lane & 15;                 // column (channel) this lane owns in B/C/D
    int hi   = lane >> 4;                 // half-wave selector
    // A is diagonal: lane only ever needs the scale for its own row m = lane&15.
    uint32_t slot_m = s0 + (uint32_t)(lane & 15);
    float scale = 0.0f;
    if (slot_m < M) scale = 1.0f / (float)cnt[uniq[slot_m]];

    v8f c = {0.f, 0.f, 0.f, 0.f, 0.f, 0.f, 0.f, 0.f};
    #pragma unroll
    for (int qk = 0; qk < 4; ++qk) {
        int k0 = qk * 4 + (hi ? 2 : 0);   // K rows held by this lane half (VGPR0)
        int k1 = k0 + 1;                  // (VGPR1)
        v2f b;
        b.x = outF[(size_t)(s0 + (uint32_t)k0) * CCH + nIdx];
        b.y = outF[(size_t)(s0 + (uint32_t)k1) * CCH + nIdx];
        v2f a;
        a.x = ((lane & 15) == k0) ? scale : 0.0f;
        a.y = ((lane & 15) == k1) ? scale : 0.0f;
        c = __builtin_amdgcn_wmma_f32_16x16x4_f32(false, a, false, b, (short)0, c, false, false);
    }
    // D (C-layout): VGPR r = row (r | r+8), lane&15 = column
    #pragma unroll
    for (int r = 0; r < 8; ++r) {
        uint32_t row = s0 + (uint32_t)r + (hi ? 8u : 0u);
        outF[(size_t)row * CCH + nIdx] = c[r];
    }
    // Coordinates: lanes 0..15 each handle one slot of the tile
    if (lane < 16) {
        uint32_t slot = s0 + (uint32_t)lane;
        int vz = -1, vy = -1, vx = -1;
        if (slot < M) {
            uint32_t key = uniq[slot];
            vz = (int)(key >> 14);
            vy = (int)((key >> 7) & 127u);
            vx = (int)(key & 127u);
        }
        outC[(size_t)slot * 3 + 0] = vz;
        outC[(size_t)slot * 3 + 1] = vy;
        outC[(size_t)slot * 3 + 2] = vx;
    }
}

extern "C" void kernel_launch(void* const* d_in, const int* in_sizes, int n_in,
                              void* d_out, int out_size, void* d_ws, size_t ws_size,
                              hipStream_t stream) {
    (void)in_sizes; (void)n_in; (void)out_size; (void)ws_size;
    const float* features = (const float*)d_in[0];
    const int*   coors    = (const int*)d_in[1];
    float* outF = (float*)d_out;                                    // [N,16] voxel_feats
    int*   outC = (int*)((char*)d_out + (size_t)NPTS * CCH * sizeof(float)); // [N,3] voxel_coors

    uint32_t* cnt  = (uint32_t*)d_ws;          // KSPACE    (8 MB)  dense per-key counts
    uint32_t* rank = cnt  + KSPACE;            // KSPACE    (8 MB)  key -> slot
    uint32_t* uniq = rank + KSPACE;            // NPTS      (8 MB)  slot -> key (sorted)
    uint32_t* bsum = uniq + NPTS;              // 1024
    uint32_t* boff = bsum + SCAN_NBLK;         // 1024
    uint32_t* mtot = boff + SCAN_NBLK;         // 1

    hipMemsetAsync(cnt,  0, (size_t)KSPACE * sizeof(uint32_t), stream);
    hipMemsetAsync(outF, 0, (size_t)NPTS * CCH * sizeof(float), stream);

    vfe_count   <<<(NPTS + 255) / 256,       256, 0, stream>>>(coors, cnt);
    vfe_scan1   <<<SCAN_NBLK,           SCAN_TPB, 0, stream>>>(cnt, rank, bsum);
    vfe_scan2   <<<1,                  SCAN_NBLK, 0, stream>>>(bsum, boff, mtot);
    vfe_compact <<<(KSPACE + 255) / 256,     256, 0, stream>>>(cnt, rank, boff, uniq);
    vfe_scatter <<<(NPTS * 4 + 255) / 256,   256, 0, stream>>>(features, coors, rank, outF);
    vfe_finalize<<<NPTS / 16 / 4,            128, 0, stream>>>(outF, outC, cnt, uniq, mtot);
}